// Push_37091337568591
// MI455X (gfx1250) — compile-verified
//
#include <hip/hip_runtime.h>
#include <stdint.h>

// Problem constants (from reference): B,C,H,W = 8,16,512,512
#define BB 8
#define CC 16
#define HH 512
#define WW 512
#define HWSZ (HH * WW)          // 262144 elements per plane
#define PIXB 256                // pixels per workgroup (1 per thread)

typedef unsigned int u32x4 __attribute__((ext_vector_type(4)));
typedef int          i32x4 __attribute__((ext_vector_type(4)));
typedef int          i32x8 __attribute__((ext_vector_type(8)));

// Non-returning L2 float atomic add. Inline asm guarantees the native
// global_atomic_add_f32 (STOREcnt path, resolved in L2 atomic units) and
// avoids any CAS-loop fallback the generic atomicAdd might pick.
__device__ __forceinline__ void gatomic_fadd(float* p, float v) {
    asm volatile("global_atomic_add_f32 %0, %1, off"
                 :
                 : "v"((unsigned long long)(uintptr_t)p), "v"(v)
                 : "memory");
}

// ---------------------------------------------------------------------------
// Scatter kernel: one workgroup = 256 consecutive pixels of one batch image.
// x tile (16 channels x 256 pixels, channel-strided in HBM) is DMA'd into LDS
// by the Tensor Data Mover; phi loads + weight math overlap with the DMA.
// ---------------------------------------------------------------------------
__global__ __launch_bounds__(256) void push_scatter(
    const float* __restrict__ x,    // [B,C,H,W]
    const float* __restrict__ phi,  // [B,2,H,W]
    float* __restrict__ out,        // [B,C,H,W] accumulator (pre-zeroed)
    float* __restrict__ cnt)        // [B,H,W]   accumulator (pre-zeroed)
{
    __shared__ float tile[CC * PIXB];   // 16 KB: tile[c*256 + t] = x[b,c,q0+t]

    const int tid  = threadIdx.x;
    const int pix0 = blockIdx.x * PIXB;       // flat over B*H*W (HW % 256 == 0)
    const int b    = pix0 / HWSZ;
    const int q0   = pix0 - b * HWSZ;

    // ---- Wave 0 issues one TDM load: [C=16 rows] x [256 elems], row stride HW.
    if (tid < 32) {
        const uint64_t ga  = (uint64_t)(uintptr_t)(x + (size_t)(b * CC) * HWSZ + q0);
        const uint32_t lds = (uint32_t)(uintptr_t)(void*)tile; // low 32b of generic ptr = LDS byte addr

        u32x4 g0;
        g0.x = 1u;                                     // count=1, user descriptor
        g0.y = lds;                                    // lds_addr
        g0.z = (uint32_t)ga;                           // global_addr[31:0]
        g0.w = (uint32_t)((ga >> 32) & 0x1FFFFFFu)     // global_addr[56:32]
             | (2u << 30);                             // type=2 ("image")

        const uint32_t td0 = HWSZ;                     // tensor_dim0 (elements)
        const uint32_t td1 = CC;                       // tensor_dim1
        const uint32_t tl0 = PIXB;                     // tile_dim0
        const uint32_t tl1 = CC;                       // tile_dim1
        const uint64_t st0 = HWSZ;                     // tensor_dim0_stride (elements)

        i32x8 g1;
        g1[0] = (int)(2u << 16);                               // data_size=4B; mask/flags=0
        g1[1] = (int)((td0 & 0xFFFFu) << 16);                  // td0[15:0]  @ bits 63:48
        g1[2] = (int)((td0 >> 16) | ((td1 & 0xFFFFu) << 16));  // td0[31:16] | td1[15:0]
        g1[3] = (int)((td1 >> 16) | (tl0 << 16));              // td1[31:16] | tile_dim0
        g1[4] = (int)(tl1);                                    // tile_dim1 | tile_dim2=0
        g1[5] = (int)(uint32_t)(st0 & 0xFFFFFFFFu);            // stride0[31:0]
        g1[6] = (int)(uint32_t)((st0 >> 32) & 0xFFFFu);        // stride0[47:32] | stride1.lo=0
        g1[7] = 0;                                             // stride1.hi

        i32x4 z4 = {0, 0, 0, 0};                // groups 2/3 unused (2-D tensor)
        i32x8 z8 = {0, 0, 0, 0, 0, 0, 0, 0};

        // amdgpu-toolchain (clang-23) 6-arg form:
        // (v4u g0, v8i g1, v4i g2, v4i g3, v8i, i32 cpol)
        __builtin_amdgcn_tensor_load_to_lds(g0, g1, z4, z4, z8, 0);
        __builtin_amdgcn_s_wait_tensorcnt(0);
    }

    // ---- Overlapped with the DMA: load phi, compute corners & weights.
    const int   q  = q0 + tid;
    const float gi = phi[(size_t)(b * 2 + 0) * HWSZ + q];
    const float gj = phi[(size_t)(b * 2 + 1) * HWSZ + q];
    const float i0f = floorf(gi), j0f = floorf(gj);
    const float wi = gi - i0f,    wj = gj - j0f;
    const int i0 = (int)i0f, j0 = (int)j0f;
    const int i0m = i0 & (HH - 1), i1m = (i0 + 1) & (HH - 1);  // circular wrap (pow2)
    const int j0m = j0 & (WW - 1), j1m = (j0 + 1) & (WW - 1);

    const float w00 = (1.0f - wi) * (1.0f - wj);
    const float w01 = (1.0f - wi) * wj;
    const float w10 = wi * (1.0f - wj);
    const float w11 = wi * wj;

    const int t00 = i0m * WW + j0m, t01 = i0m * WW + j1m;
    const int t10 = i1m * WW + j0m, t11 = i1m * WW + j1m;

    // Count scatter (independent of x tile -> do it before the barrier).
    float* cb = cnt + (size_t)b * HWSZ;
    gatomic_fadd(cb + t00, w00);
    gatomic_fadd(cb + t01, w01);
    gatomic_fadd(cb + t10, w10);
    gatomic_fadd(cb + t11, w11);

    __syncthreads();   // TDM data now visible to all 8 waves

    // Channel scatter: 16 ds reads (conflict-free: bank = tid % 64) + 64 atomics.
    #pragma unroll
    for (int c = 0; c < CC; ++c) {
        const float v = tile[c * PIXB + tid];
        float* ob = out + (size_t)(b * CC + c) * HWSZ;
        gatomic_fadd(ob + t00, v * w00);
        gatomic_fadd(ob + t01, v * w01);
        gatomic_fadd(ob + t10, v * w10);
        gatomic_fadd(ob + t11, v * w11);
    }
}

// ---------------------------------------------------------------------------
// Normalize: out[b,c,i,j] /= max(cnt[b,i,j], 1e-3). float4 streaming pass.
// ---------------------------------------------------------------------------
__global__ __launch_bounds__(256) void push_normalize(
    float* __restrict__ out, const float* __restrict__ cnt)
{
    const size_t g    = (size_t)blockIdx.x * 256 + threadIdx.x; // float4 group id
    const size_t base = g * 4;                                  // flat into [B,C,H,W]
    const size_t plane = base / HWSZ;                           // b*C + c
    const size_t q     = base - plane * HWSZ;                   // multiple of 4
    const size_t b     = plane / CC;

    const float4 cv = *(const float4*)(cnt + b * HWSZ + q);
    float4 ov = *(float4*)(out + base);
    ov.x /= fmaxf(cv.x, 0.001f);
    ov.y /= fmaxf(cv.y, 0.001f);
    ov.z /= fmaxf(cv.z, 0.001f);
    ov.w /= fmaxf(cv.w, 0.001f);
    *(float4*)(out + base) = ov;
}

extern "C" void kernel_launch(void* const* d_in, const int* in_sizes, int n_in,
                              void* d_out, int out_size, void* d_ws, size_t ws_size,
                              hipStream_t stream) {
    const float* x   = (const float*)d_in[0];   // [B,C,H,W]
    const float* phi = (const float*)d_in[1];   // [B,2,H,W]
    float* out = (float*)d_out;                 // [B,C,H,W]
    float* cnt = (float*)d_ws;                  // [B,H,W] scratch

    // Accumulators must be zero every launch (harness does not re-poison).
    (void)hipMemsetAsync(out, 0, (size_t)BB * CC * HWSZ * sizeof(float), stream);
    (void)hipMemsetAsync(cnt, 0, (size_t)BB * HWSZ * sizeof(float), stream);

    const int scatter_blocks = (BB * HWSZ) / PIXB;              // 8192
    push_scatter<<<scatter_blocks, 256, 0, stream>>>(x, phi, out, cnt);

    const int norm_blocks = (int)(((size_t)BB * CC * HWSZ / 4) / 256); // 32768
    push_normalize<<<norm_blocks, 256, 0, stream>>>(out, cnt);
}